// GATClassifier_77747497992892
// MI455X (gfx1250) — compile-verified
//
#include <hip/hip_runtime.h>
#include <hip/hip_bf16.h>
#include <math.h>

#define NNODES 50000
#define NFEAT  128
#define NEDGE  800000
#define NTOT   (NEDGE + NNODES)
#define NEG_SLOPE 0.2f

typedef float v2f __attribute__((ext_vector_type(2)));
typedef float v8f __attribute__((ext_vector_type(8)));

// Order-preserving float -> uint encoding so integer atomicMax == float max.
__device__ __forceinline__ unsigned fenc(float f) {
    unsigned u = __float_as_uint(f);
    return (u & 0x80000000u) ? ~u : (u | 0x80000000u);
}
__device__ __forceinline__ float fdec(unsigned k) {
    return (k & 0x80000000u) ? __uint_as_float(k ^ 0x80000000u)
                             : __uint_as_float(~k);
}

// ---------------------------------------------------------------------------
// Dual GEMM: xl = X @ Wl + bl ; xr = X @ Wr + br   (fp32 WMMA 16x16x4)
// blockIdx.x = row_tile*2 + which;  256 threads = 8 waves, wave w -> col tile w
// ---------------------------------------------------------------------------
__global__ __launch_bounds__(256) void gemm_kernel(
    const float* __restrict__ X,
    const float* __restrict__ Wl, const float* __restrict__ bl,
    const float* __restrict__ Wr, const float* __restrict__ br,
    float* __restrict__ xl, float* __restrict__ xr)
{
    __shared__ float Atile[16 * 132];   // padded row stride -> no bank conflicts

    const int rt    = blockIdx.x >> 1;
    const int which = blockIdx.x & 1;
    const float* __restrict__ W    = which ? Wr : Wl;
    const float* __restrict__ bias = which ? br : bl;
    float* __restrict__ out        = which ? xr : xl;

    const int row0 = rt * 16;
    const int tid  = threadIdx.x;

    // Stage X[row0..row0+15][0..127] into LDS: 256 threads x 8 floats
    {
        const int r  = tid >> 4;        // 0..15
        const int c8 = (tid & 15) * 8;  // 0..120
        const float4* src = (const float4*)(X + (size_t)(row0 + r) * NFEAT + c8);
        float4 v0 = src[0];
        float4 v1 = src[1];
        float* drow = &Atile[r * 132 + c8];
        *(float4*)(drow)     = v0;
        *(float4*)(drow + 4) = v1;
    }
    __syncthreads();

    const int wave  = tid >> 5;          // 0..7 -> column tile
    const int lane  = tid & 31;
    const int laneN = lane & 15;         // M (for A) / N (for B,C)
    const int hi    = lane >> 4;         // selects K offset 0 or 2
    const int n0    = wave * 16;

    v8f c = {};
    #pragma unroll
    for (int k = 0; k < NFEAT; k += 4) {
        const int ka = k + hi * 2;
        v2f a;
        a.x = Atile[laneN * 132 + ka];
        a.y = Atile[laneN * 132 + ka + 1];
        v2f b;
        b.x = W[(size_t)ka * NFEAT + n0 + laneN];
        b.y = W[(size_t)(ka + 1) * NFEAT + n0 + laneN];
        c = __builtin_amdgcn_wmma_f32_16x16x4_f32(
                false, a, false, b, (short)0, c, false, false);
    }

    const float bcol = bias[n0 + laneN];
    #pragma unroll
    for (int v = 0; v < 8; ++v) {
        const int row = row0 + v + hi * 8;
        out[(size_t)row * NFEAT + n0 + laneN] = c[v] + bcol;
    }
}

// ---------------------------------------------------------------------------
// Zero h accumulator, asum; init encoded emax to minimal key
// ---------------------------------------------------------------------------
__global__ void init_kernel(float* __restrict__ h,
                            unsigned* __restrict__ emax,
                            float* __restrict__ asum)
{
    const int i = blockIdx.x * blockDim.x + threadIdx.x;
    if (i < NNODES * NFEAT) h[i] = 0.0f;
    if (i < NNODES) { emax[i] = 0u; asum[i] = 0.0f; }
}

// ---------------------------------------------------------------------------
// e_k = att . leaky_relu(xl[src] + xr[dst]) ; segment max (encoded atomic)
// one wave32 per edge, lane handles 4 features (float4)
// ---------------------------------------------------------------------------
__global__ __launch_bounds__(256) void edge_e_kernel(
    const int* __restrict__ ei,
    const float* __restrict__ xl, const float* __restrict__ xr,
    const float* __restrict__ att,
    float* __restrict__ ebuf, unsigned* __restrict__ emax)
{
    const int edge = blockIdx.x * 8 + (threadIdx.x >> 5);
    if (edge >= NTOT) return;
    const int lane = threadIdx.x & 31;

    int s, d;
    if (edge < NEDGE) { s = ei[edge]; d = ei[NEDGE + edge]; }
    else              { s = edge - NEDGE; d = s; }

    const float4 a = *(const float4*)(xl + (size_t)s * NFEAT + lane * 4);
    const float4 b = *(const float4*)(xr + (size_t)d * NFEAT + lane * 4);
    const float4 t = *(const float4*)(att + lane * 4);

    float acc = 0.0f, v;
    v = a.x + b.x; v = v > 0.0f ? v : NEG_SLOPE * v; acc += t.x * v;
    v = a.y + b.y; v = v > 0.0f ? v : NEG_SLOPE * v; acc += t.y * v;
    v = a.z + b.z; v = v > 0.0f ? v : NEG_SLOPE * v; acc += t.z * v;
    v = a.w + b.w; v = v > 0.0f ? v : NEG_SLOPE * v; acc += t.w * v;

    #pragma unroll
    for (int off = 16; off > 0; off >>= 1)
        acc += __shfl_down(acc, off, 32);

    if (lane == 0) {
        ebuf[edge] = acc;
        atomicMax(&emax[d], fenc(acc));
    }
}

// ---------------------------------------------------------------------------
// a_k = exp(e_k - emax[dst]) ; segment sum of a_k
// ---------------------------------------------------------------------------
__global__ void edge_norm_kernel(const int* __restrict__ ei,
                                 float* __restrict__ ebuf,
                                 const unsigned* __restrict__ emax,
                                 float* __restrict__ asum)
{
    const int idx = blockIdx.x * blockDim.x + threadIdx.x;
    if (idx >= NTOT) return;
    const int d = (idx < NEDGE) ? ei[NEDGE + idx] : (idx - NEDGE);
    const float m = fdec(emax[d]);
    const float a = __expf(ebuf[idx] - m);
    ebuf[idx] = a;
    unsafeAtomicAdd(&asum[d], a);
}

// ---------------------------------------------------------------------------
// h[dst] += (a_k / asum[dst]) * xl[src]   (4 hw fp32 atomics per lane)
// ---------------------------------------------------------------------------
__global__ __launch_bounds__(256) void edge_scatter_kernel(
    const int* __restrict__ ei,
    const float* __restrict__ xl,
    const float* __restrict__ ebuf, const float* __restrict__ asum,
    float* __restrict__ h)
{
    const int edge = blockIdx.x * 8 + (threadIdx.x >> 5);
    if (edge >= NTOT) return;
    const int lane = threadIdx.x & 31;

    int s, d;
    if (edge < NEDGE) { s = ei[edge]; d = ei[NEDGE + edge]; }
    else              { s = edge - NEDGE; d = s; }

    const float w = ebuf[edge] / asum[d];
    const float4 v = *(const float4*)(xl + (size_t)s * NFEAT + lane * 4);
    float* dst = h + (size_t)d * NFEAT + lane * 4;
    unsafeAtomicAdd(dst + 0, w * v.x);
    unsafeAtomicAdd(dst + 1, w * v.y);
    unsafeAtomicAdd(dst + 2, w * v.z);
    unsafeAtomicAdd(dst + 3, w * v.w);
}

// ---------------------------------------------------------------------------
// out[i] = sigmoid((h[i] + bias) . Wo + bo)   one wave per node
// ---------------------------------------------------------------------------
__global__ __launch_bounds__(256) void out_kernel(
    const float* __restrict__ h, const float* __restrict__ bias,
    const float* __restrict__ Wo, const float* __restrict__ bo,
    float* __restrict__ out)
{
    const int node = blockIdx.x * 8 + (threadIdx.x >> 5);
    if (node >= NNODES) return;
    const int lane = threadIdx.x & 31;

    const float4 hv = *(const float4*)(h + (size_t)node * NFEAT + lane * 4);
    const float4 bv = *(const float4*)(bias + lane * 4);
    const float4 wv = *(const float4*)(Wo + lane * 4);

    float acc = (hv.x + bv.x) * wv.x + (hv.y + bv.y) * wv.y +
                (hv.z + bv.z) * wv.z + (hv.w + bv.w) * wv.w;

    #pragma unroll
    for (int off = 16; off > 0; off >>= 1)
        acc += __shfl_down(acc, off, 32);

    if (lane == 0) {
        const float z = acc + bo[0];
        out[node] = 1.0f / (1.0f + __expf(-z));
    }
}

// ---------------------------------------------------------------------------
extern "C" void kernel_launch(void* const* d_in, const int* in_sizes, int n_in,
                              void* d_out, int out_size, void* d_ws, size_t ws_size,
                              hipStream_t stream)
{
    const float* x    = (const float*)d_in[0];
    const int*   ei   = (const int*)d_in[1];
    const float* Wl   = (const float*)d_in[2];
    const float* bl   = (const float*)d_in[3];
    const float* Wr   = (const float*)d_in[4];
    const float* br   = (const float*)d_in[5];
    const float* att  = (const float*)d_in[6];
    const float* bias = (const float*)d_in[7];
    const float* Wo   = (const float*)d_in[8];
    const float* bo   = (const float*)d_in[9];
    float* out = (float*)d_out;

    // Workspace layout (all 16B aligned)
    float* xl      = (float*)d_ws;                    // N*F
    float* xr      = xl + (size_t)NNODES * NFEAT;     // N*F
    float* h       = xr + (size_t)NNODES * NFEAT;     // N*F
    float* ebuf    = h  + (size_t)NNODES * NFEAT;     // NTOT
    unsigned* emax = (unsigned*)(ebuf + NTOT);        // N
    float* asum    = (float*)(emax + NNODES);         // N

    // 1) dual GEMM (WMMA f32): 3125 row tiles x 2 matrices
    gemm_kernel<<<dim3(3125 * 2), dim3(256), 0, stream>>>(
        x, Wl, bl, Wr, br, xl, xr);

    // 2) init accumulators
    init_kernel<<<dim3((NNODES * NFEAT + 255) / 256), dim3(256), 0, stream>>>(
        h, emax, asum);

    // 3) edge scores + segment max
    edge_e_kernel<<<dim3((NTOT + 7) / 8), dim3(256), 0, stream>>>(
        ei, xl, xr, att, ebuf, emax);

    // 4) exp + segment sum
    edge_norm_kernel<<<dim3((NTOT + 255) / 256), dim3(256), 0, stream>>>(
        ei, ebuf, emax, asum);

    // 5) weighted scatter
    edge_scatter_kernel<<<dim3((NTOT + 7) / 8), dim3(256), 0, stream>>>(
        ei, xl, ebuf, asum, h);

    // 6) head + sigmoid
    out_kernel<<<dim3((NNODES + 7) / 8), dim3(256), 0, stream>>>(
        h, bias, Wo, bo, out);
}